// ChamferLoss_71322226917415
// MI455X (gfx1250) — compile-verified
//
#include <hip/hip_runtime.h>
#include <math.h>

typedef __attribute__((ext_vector_type(2))) float v2f;
typedef __attribute__((ext_vector_type(8))) float v8f;

#define TILE 16
#define WAVES 8
#define THREADS 256
#define UNROLL 4   // tiles per wave per iteration (pairs share a min3)

// Single-instruction float min (no canonicalize): acc = min_num(acc, x).
__device__ __forceinline__ void fmin1(float& acc, float x) {
    asm("v_min_num_f32 %0, %0, %1" : "+v"(acc) : "v"(x));
}

// Three-operand min: acc = min_num(acc, x, y) in one VALU instruction.
__device__ __forceinline__ void fmin3(float& acc, float x, float y) {
    asm("v_min3_num_f32 %0, %0, %1, %2" : "+v"(acc) : "v"(x), "v"(y));
}

// ---------------------------------------------------------------------------
// Prologue: homogeneous fragment forms for a point tensor P (npts, 3):
//   formA[i] = ( x,  y,  z, 1 )          -> A-matrix rows [x y z 1]
//   formB[i] = (-2x,-2y,-2z, ||p||^2 )   -> B-matrix cols [-2p ; ||p||^2]
// One V_WMMA_F32_16X16X4_F32 then computes t = ||y||^2 - 2 x.y directly.
// ---------------------------------------------------------------------------
__global__ void chamfer_transform_kernel(const float* __restrict__ P, int npts,
                                         float4* __restrict__ formA,
                                         float4* __restrict__ formB)
{
    int i = blockIdx.x * blockDim.x + threadIdx.x;
    if (i < npts) {
        const float x = P[3 * i + 0];
        const float y = P[3 * i + 1];
        const float z = P[3 * i + 2];
        formA[i] = make_float4(x, y, z, 1.0f);
        formB[i] = make_float4(-2.0f * x, -2.0f * y, -2.0f * z,
                               x * x + y * y + z * z);
    }
}

// ---------------------------------------------------------------------------
// Fast directed pass. Block = one 16-row strip of X in one batch; 8 waves
// split Y tiles. Per iteration: 4 b64 loads issued up-front (immediate
// offsets off one running pointer), then 4 wmma; WMMA results folded into
// the accumulators pairwise with v_min3_num_f32 (4 VALU per wmma).
// ---------------------------------------------------------------------------
__global__ __launch_bounds__(THREADS) void chamfer_dir_fast(
    const float4* __restrict__ XA,   // A-form of X (x,y,z,1)
    const float4* __restrict__ XB,   // B-form of X (.w = ||x||^2)
    const float4* __restrict__ YB,   // B-form of Y (-2y, ||y||^2)
    int Nx, int Ny, int strips, float* __restrict__ strip_sums)
{
    const int strip = blockIdx.x;
    const int b     = blockIdx.y;
    const int lane  = threadIdx.x & 31;
    const int wave  = threadIdx.x >> 5;
    const int h     = lane >> 4;     // lane half: K pair {0,1} vs {2,3}; row v vs v+8
    const int l16   = lane & 15;
    const int n0    = strip * TILE;

    const float4* __restrict__ XAb = XA + (size_t)b * Nx;
    const float4* __restrict__ XBb = XB + (size_t)b * Nx;
    const float4* __restrict__ YBb = YB + (size_t)b * Ny;

    // A fragment: (K0,K1)=(x,y) for h=0; (K2,K3)=(z,1) for h=1 -> one b64 load
    const v2f afrag = ((const v2f*)(XAb + n0 + l16))[h];

    const float INF = __builtin_inff();
    float rm[2][8];                  // two accumulator sets (pairs of tiles)
#pragma unroll
    for (int u = 0; u < 2; ++u)
#pragma unroll
        for (int v = 0; v < 8; ++v) rm[u][v] = INF;

    const int ntiles = Ny / TILE;                    // 512 for 8192 points
    const int iters  = ntiles / (WAVES * UNROLL);    // 16
    // One tile = TILE * sizeof(float4) = 256 B. Unroll stride = WAVES tiles
    // = 2048 B (immediate). Loop stride = WAVES*UNROLL tiles = 8192 B.
    const char* bp = (const char*)YBb
                   + (size_t)(wave * TILE + l16) * sizeof(float4)
                   + (size_t)h * 8;

    for (int it = 0; it < iters; ++it) {
        // Issue all loads first: 4 outstanding b64 loads per wave.
        v2f bf0 = *(const v2f*)(bp + 0 * 2048);
        v2f bf1 = *(const v2f*)(bp + 1 * 2048);
        v2f bf2 = *(const v2f*)(bp + 2 * 2048);
        v2f bf3 = *(const v2f*)(bp + 3 * 2048);
        bp += WAVES * UNROLL * TILE * sizeof(float4);

        const v8f c = {0.f, 0.f, 0.f, 0.f, 0.f, 0.f, 0.f, 0.f};
        v8f d0 = __builtin_amdgcn_wmma_f32_16x16x4_f32(
            false, afrag, false, bf0, (short)0, c, false, false);
        v8f d1 = __builtin_amdgcn_wmma_f32_16x16x4_f32(
            false, afrag, false, bf1, (short)0, c, false, false);
#pragma unroll
        for (int v = 0; v < 8; ++v) fmin3(rm[0][v], d0[v], d1[v]);

        v8f d2 = __builtin_amdgcn_wmma_f32_16x16x4_f32(
            false, afrag, false, bf2, (short)0, c, false, false);
        v8f d3 = __builtin_amdgcn_wmma_f32_16x16x4_f32(
            false, afrag, false, bf3, (short)0, c, false, false);
#pragma unroll
        for (int v = 0; v < 8; ++v) fmin3(rm[1][v], d2[v], d3[v]);
    }

    // Tail (not taken for Ny multiple of 512; kept for generality).
    for (int t = iters * WAVES * UNROLL + wave; t < ntiles; t += WAVES) {
        const v2f bf = ((const v2f*)(YBb + t * TILE + l16))[h];
        const v8f c = {0.f, 0.f, 0.f, 0.f, 0.f, 0.f, 0.f, 0.f};
        v8f d = __builtin_amdgcn_wmma_f32_16x16x4_f32(
            false, afrag, false, bf, (short)0, c, false, false);
#pragma unroll
        for (int v = 0; v < 8; ++v) fmin1(rm[0][v], d[v]);
    }

    // Merge sets, butterfly min within each 16-lane half (wave32).
#pragma unroll
    for (int v = 0; v < 8; ++v) {
        float r = fminf(rm[0][v], rm[1][v]);
#pragma unroll
        for (int off = 1; off < 16; off <<= 1)
            r = fminf(r, __shfl_xor(r, off, 32));
        rm[0][v] = r;                // min of t for row (v + 8h) over this wave
    }

    __shared__ float lmin[WAVES][TILE];
    __shared__ float rowval[TILE];
    if (l16 == 0) {
#pragma unroll
        for (int v = 0; v < 8; ++v)
            lmin[wave][v + 8 * h] = rm[0][v];
    }
    __syncthreads();

    // Combine waves, re-add x^2 (from B-form .w), apply the hoisted clamp.
    if (threadIdx.x < TILE) {
        const int r = threadIdx.x;
        float t = lmin[0][r];
#pragma unroll
        for (int w = 1; w < WAVES; ++w) t = fminf(t, lmin[w][r]);
        const float x2 = XBb[n0 + r].w;
        rowval[r] = fmaxf(t + x2, 0.0f);
    }
    __syncthreads();

    if (threadIdx.x == 0) {
        float s = 0.0f;
#pragma unroll
        for (int r = 0; r < TILE; ++r) s += rowval[r];
        strip_sums[(size_t)b * strips + strip] = s;
    }
}

// ---------------------------------------------------------------------------
// Fallback (self-contained, raw f32 inputs) if workspace is too small.
// ---------------------------------------------------------------------------
__global__ __launch_bounds__(THREADS) void chamfer_dir_slow(
    const float* __restrict__ X, const float* __restrict__ Y,
    int Nx, int Ny, int strips, float* __restrict__ strip_sums)
{
    const int strip = blockIdx.x;
    const int b     = blockIdx.y;
    const int lane  = threadIdx.x & 31;
    const int wave  = threadIdx.x >> 5;
    const int h     = lane >> 4;
    const int l16   = lane & 15;
    const int n0    = strip * TILE;

    const float* __restrict__ Xb = X + (size_t)b * Nx * 3;
    const float* __restrict__ Yb = Y + (size_t)b * Ny * 3;

    const float* xp = Xb + (size_t)(n0 + l16) * 3;
    const float ax = xp[0], ay = xp[1], az = xp[2];
    v2f afrag;
    afrag.x = h ? az : ax;
    afrag.y = h ? 1.0f : ay;         // homogeneous: K3 = 1

    const float INF = __builtin_inff();
    float rm0[8], rm1[8];
#pragma unroll
    for (int v = 0; v < 8; ++v) { rm0[v] = INF; rm1[v] = INF; }

    const int ntiles = Ny / TILE;
    for (int t = wave; t < ntiles; t += WAVES * 2) {
        {
            const int col = t * TILE + l16;
            const float* yp = Yb + (size_t)col * 3;
            const float yx = yp[0], yy = yp[1], yz = yp[2];
            const float y2 = yx * yx + yy * yy + yz * yz;
            v2f bfrag;
            bfrag.x = h ? (-2.0f * yz) : (-2.0f * yx);
            bfrag.y = h ? y2 : (-2.0f * yy);
            const v8f c = {0.f, 0.f, 0.f, 0.f, 0.f, 0.f, 0.f, 0.f};
            v8f d = __builtin_amdgcn_wmma_f32_16x16x4_f32(
                false, afrag, false, bfrag, (short)0, c, false, false);
#pragma unroll
            for (int v = 0; v < 8; ++v) fmin1(rm0[v], d[v]);
        }
        {
            const int col = (t + WAVES) * TILE + l16;
            const float* yp = Yb + (size_t)col * 3;
            const float yx = yp[0], yy = yp[1], yz = yp[2];
            const float y2 = yx * yx + yy * yy + yz * yz;
            v2f bfrag;
            bfrag.x = h ? (-2.0f * yz) : (-2.0f * yx);
            bfrag.y = h ? y2 : (-2.0f * yy);
            const v8f c = {0.f, 0.f, 0.f, 0.f, 0.f, 0.f, 0.f, 0.f};
            v8f d = __builtin_amdgcn_wmma_f32_16x16x4_f32(
                false, afrag, false, bfrag, (short)0, c, false, false);
#pragma unroll
            for (int v = 0; v < 8; ++v) fmin1(rm1[v], d[v]);
        }
    }

#pragma unroll
    for (int v = 0; v < 8; ++v) {
        float r = fminf(rm0[v], rm1[v]);
#pragma unroll
        for (int off = 1; off < 16; off <<= 1)
            r = fminf(r, __shfl_xor(r, off, 32));
        rm0[v] = r;
    }

    __shared__ float lmin[WAVES][TILE];
    __shared__ float rowval[TILE];
    if (l16 == 0) {
#pragma unroll
        for (int v = 0; v < 8; ++v)
            lmin[wave][v + 8 * h] = rm0[v];
    }
    __syncthreads();

    if (threadIdx.x < TILE) {
        const int r = threadIdx.x;
        float t = lmin[0][r];
#pragma unroll
        for (int w = 1; w < WAVES; ++w) t = fminf(t, lmin[w][r]);
        const float* xr = Xb + (size_t)(n0 + r) * 3;
        const float x2 = xr[0] * xr[0] + xr[1] * xr[1] + xr[2] * xr[2];
        rowval[r] = fmaxf(t + x2, 0.0f);
    }
    __syncthreads();

    if (threadIdx.x == 0) {
        float s = 0.0f;
#pragma unroll
        for (int r = 0; r < TILE; ++r) s += rowval[r];
        strip_sums[(size_t)b * strips + strip] = s;
    }
}

// Deterministic final reduction: fixed-order strip sums, means, per-batch max,
// batch mean.
__global__ void chamfer_finalize(const float* __restrict__ rec_sums,
                                 const float* __restrict__ data_sums,
                                 int strips_rec, int strips_data,
                                 int N, int M, int B, float* __restrict__ out)
{
    __shared__ float sums[2][8];
    const int tid = threadIdx.x;
    if (tid < 2 * B) {
        const int dir = tid & 1;
        const int b   = tid >> 1;
        const float* p   = dir ? data_sums : rec_sums;
        const int strips = dir ? strips_data : strips_rec;
        const float cnt  = dir ? (float)M : (float)N;
        float s = 0.0f;
        for (int i = 0; i < strips; ++i) s += p[(size_t)b * strips + i];
        sums[dir][b] = s / cnt;
    }
    __syncthreads();
    if (tid == 0) {
        float acc = 0.0f;
        for (int b = 0; b < B; ++b)
            acc += fmaxf(sums[0][b], sums[1][b]);
        out[0] = acc / (float)B;
    }
}

extern "C" void kernel_launch(void* const* d_in, const int* in_sizes, int n_in,
                              void* d_out, int out_size, void* d_ws, size_t ws_size,
                              hipStream_t stream)
{
    const float* rec  = (const float*)d_in[0];   // (B, N, 3) f32
    const float* data = (const float*)d_in[1];   // (B, M, 3) f32
    const int B = 4;
    const int N = in_sizes[0] / (B * 3);
    const int M = in_sizes[1] / (B * 3);
    const int strips_rec  = N / TILE;
    const int strips_data = M / TILE;
    const size_t nrec  = (size_t)B * N;
    const size_t ndata = (size_t)B * M;

    const size_t sums_floats = (size_t)B * (strips_rec + strips_data);
    const size_t need_fast = (nrec + ndata) * 2 * sizeof(float4)
                           + sums_floats * sizeof(float);

    dim3 blk(THREADS);

    if (ws_size >= need_fast) {
        // Workspace layout: recA | recB | dataA | dataB | strip sums
        float4* recA  = (float4*)d_ws;
        float4* recB  = recA + nrec;
        float4* dataA = recB + nrec;
        float4* dataB = dataA + ndata;
        float*  sums  = (float*)(dataB + ndata);
        float*  rec_sums  = sums;
        float*  data_sums = sums + (size_t)B * strips_rec;

        chamfer_transform_kernel<<<(int)((nrec + 255) / 256), 256, 0, stream>>>(
            rec, (int)nrec, recA, recB);
        chamfer_transform_kernel<<<(int)((ndata + 255) / 256), 256, 0, stream>>>(
            data, (int)ndata, dataA, dataB);

        chamfer_dir_fast<<<dim3(strips_rec, B), blk, 0, stream>>>(
            recA, recB, dataB, N, M, strips_rec, rec_sums);
        chamfer_dir_fast<<<dim3(strips_data, B), blk, 0, stream>>>(
            dataA, dataB, recB, M, N, strips_data, data_sums);

        chamfer_finalize<<<1, 64, 0, stream>>>(rec_sums, data_sums,
                                               strips_rec, strips_data,
                                               N, M, B, (float*)d_out);
    } else {
        float* rec_sums  = (float*)d_ws;
        float* data_sums = rec_sums + (size_t)B * strips_rec;

        chamfer_dir_slow<<<dim3(strips_rec, B), blk, 0, stream>>>(
            rec, data, N, M, strips_rec, rec_sums);
        chamfer_dir_slow<<<dim3(strips_data, B), blk, 0, stream>>>(
            data, rec, M, N, strips_data, data_sums);

        chamfer_finalize<<<1, 64, 0, stream>>>(rec_sums, data_sums,
                                               strips_rec, strips_data,
                                               N, M, B, (float*)d_out);
    }
}